// TransformerBlockRes_40312563040382
// MI455X (gfx1250) — compile-verified
//
#include <hip/hip_runtime.h>
#include <hip/hip_bf16.h>

#define BATCH 16
#define NPTS  2048
#define KNN   16
#define KP1   17
#define CH    64
#define ES    65      // sE row stride (floats): conflict-free A-fragment column reads
#define DVS   64      // sDV row stride: only ever row-fixed, lane-varying accesses
#define PS    72      // packed-W pair-row stride (in v2f pairs): halves 16 banks apart

typedef __attribute__((ext_vector_type(2))) float v2f;
typedef __attribute__((ext_vector_type(8))) float v8f;

// ---------------------------------------------------------------------------
// Kernel 1: exact per-cloud KNN (top-16 nearest, self excluded, self appended)
// ---------------------------------------------------------------------------
__global__ __launch_bounds__(256) void knn_kernel(const float* __restrict__ pos,
                                                  int* __restrict__ nbr) {
  __shared__ float sp[256 * 3];
  const int cloud = blockIdx.x >> 3;                       // 8 blocks per cloud
  const int i_loc = ((blockIdx.x & 7) << 8) + threadIdx.x; // target within cloud
  const int gi    = cloud * NPTS + i_loc;
  const float px = pos[gi * 3 + 0];
  const float py = pos[gi * 3 + 1];
  const float pz = pos[gi * 3 + 2];

  float bd[KNN];
  int   bi[KNN];
#pragma unroll
  for (int t = 0; t < KNN; ++t) { bd[t] = 3.0e38f; bi[t] = 0; }

  for (int tile = 0; tile < NPTS / 256; ++tile) {
    const int cj = cloud * NPTS + tile * 256 + threadIdx.x;
    sp[threadIdx.x * 3 + 0] = pos[cj * 3 + 0];
    sp[threadIdx.x * 3 + 1] = pos[cj * 3 + 1];
    sp[threadIdx.x * 3 + 2] = pos[cj * 3 + 2];
    __syncthreads();
    const int jbase = tile * 256;
#pragma unroll 4
    for (int jj = 0; jj < 256; ++jj) {
      const int j = jbase + jj;
      const float dx = px - sp[jj * 3 + 0];
      const float dy = py - sp[jj * 3 + 1];
      const float dz = pz - sp[jj * 3 + 2];
      float d2 = dx * dx + dy * dy + dz * dz;
      if (j == i_loc) d2 = 3.9e38f;                        // exclude self
      if (d2 < bd[KNN - 1]) {
#pragma unroll
        for (int t = KNN - 1; t >= 0; --t) {               // sorted insert, static idx
          const bool shift = (t > 0) && (d2 < bd[(t > 0) ? (t - 1) : 0]);
          const bool place = (d2 < bd[t]) && !shift;
          const float nd = shift ? bd[(t > 0) ? (t - 1) : 0] : (place ? d2 : bd[t]);
          const int   ni = shift ? bi[(t > 0) ? (t - 1) : 0] : (place ? j  : bi[t]);
          bd[t] = nd; bi[t] = ni;
        }
      }
    }
    __syncthreads();
  }
#pragma unroll
  for (int t = 0; t < KNN; ++t) nbr[gi * KP1 + t] = bi[t];
  nbr[gi * KP1 + KNN] = i_loc;                             // self loop last (ref order)
}

// ---------------------------------------------------------------------------
// Kernel 2: fused projections  Y = X @ W  for W_dst / W_src / W_val (fp32 WMMA)
// Weights pre-packed in LDS in B-fragment order: (W[2p][c], W[2p+1][c]) pairs
// contiguous -> each B fragment is one aligned ds_load_b64, no packing movs.
// ---------------------------------------------------------------------------
__global__ __launch_bounds__(256) void proj_kernel(const float* __restrict__ X,
                                                   const float* __restrict__ Wdst,
                                                   const float* __restrict__ Wsrc,
                                                   const float* __restrict__ Wval,
                                                   float* __restrict__ Adst,
                                                   float* __restrict__ Asrc,
                                                   float* __restrict__ Vp) {
  __shared__ float sW[32 * PS * 2];                        // 32 k-pairs x PS cols x 2
  const float* W = (blockIdx.y == 0) ? Wdst : (blockIdx.y == 1) ? Wsrc : Wval;
  float*       Y = (blockIdx.y == 0) ? Adst : (blockIdx.y == 1) ? Asrc : Vp;
  for (int idx = threadIdx.x; idx < CH * CH; idx += 256) {
    const int k = idx >> 6, c = idx & 63;
    sW[(((k >> 1) * PS) + c) * 2 + (k & 1)] = W[idx];
  }
  __syncthreads();

  const int wave = threadIdx.x >> 5;
  const int lane = threadIdx.x & 31;
  const int half = lane >> 4;                              // K sub-pair select
  const int l16  = lane & 15;
  const int row  = blockIdx.x * 128 + wave * 16 + l16;     // A-fragment row
  const float* xr = X + (size_t)row * CH;

  v8f acc0 = {}, acc1 = {}, acc2 = {}, acc3 = {};
#pragma unroll
  for (int kk = 0; kk < CH; kk += 4) {
    const int kb = kk + 2 * half;
    v2f a;  a.x = xr[kb];  a.y = xr[kb + 1];               // global_load_b64
    const int pr = (kb >> 1) * PS;                         // packed pair-row base
    const v2f b0 = *(const v2f*)&sW[(pr + l16 +  0) * 2];  // ds_load_b64 each
    const v2f b1 = *(const v2f*)&sW[(pr + l16 + 16) * 2];
    const v2f b2 = *(const v2f*)&sW[(pr + l16 + 32) * 2];
    const v2f b3 = *(const v2f*)&sW[(pr + l16 + 48) * 2];
    acc0 = __builtin_amdgcn_wmma_f32_16x16x4_f32(false, a, false, b0, (short)0, acc0, false, false);
    acc1 = __builtin_amdgcn_wmma_f32_16x16x4_f32(false, a, false, b1, (short)0, acc1, false, false);
    acc2 = __builtin_amdgcn_wmma_f32_16x16x4_f32(false, a, false, b2, (short)0, acc2, false, false);
    acc3 = __builtin_amdgcn_wmma_f32_16x16x4_f32(false, a, false, b3, (short)0, acc3, false, false);
  }
#pragma unroll
  for (int v = 0; v < 8; ++v) {                            // C/D layout: M=v(+8), N=lane%16
    const int orow = blockIdx.x * 128 + wave * 16 + v + 8 * half;
    Y[(size_t)orow * CH +  0 + l16] = acc0[v];
    Y[(size_t)orow * CH + 16 + l16] = acc1[v];
    Y[(size_t)orow * CH + 32 + l16] = acc2[v];
    Y[(size_t)orow * CH + 48 + l16] = acc3[v];
  }
}

// ---------------------------------------------------------------------------
// Kernel 3: edge features -> attn GEMM (WMMA) -> 17-way softmax -> aggregate
// 16 targets/block => 272 edges = exactly 17 M-tiles of 16.
// LDS = 160,064 B  (two workgroups fit in a 320KB WGP).
// ---------------------------------------------------------------------------
__global__ __launch_bounds__(256) void edge_kernel(const float* __restrict__ pos,
                                                   const int* __restrict__ nbr,
                                                   const float* __restrict__ Adst,
                                                   const float* __restrict__ Asrc,
                                                   const float* __restrict__ Vp,
                                                   const float* __restrict__ Wp,
                                                   const float* __restrict__ bp,
                                                   const float* __restrict__ Wa,
                                                   const float* __restrict__ ba,
                                                   float* __restrict__ out) {
  extern __shared__ float sm[];
  float* sE  = sm;                    // 272 x 65  (edge features, then scores in-place)
  float* sDV = sE  + 272 * ES;        // 272 x 64  (val_j + delta)
  float* sWa = sDV + 272 * DVS;       // packed 32 x PS x 2
  float* sWp = sWa + 32 * PS * 2;     // 3 x 64
  float* sbp = sWp + 192;             // 64
  float* sba = sbp + 64;              // 64

  const int tid = threadIdx.x;
  for (int idx = tid; idx < CH * CH; idx += 256) {
    const int k = idx >> 6, c = idx & 63;
    sWa[(((k >> 1) * PS) + c) * 2 + (k & 1)] = Wa[idx];    // fragment-order pack
  }
  if (tid < 192) sWp[tid] = Wp[tid];
  if (tid < 64)  { sbp[tid] = bp[tid]; sba[tid] = ba[tid]; }

  const int base_t = blockIdx.x * 16;               // 16 targets, all in one cloud
  const int cbase  = (base_t / NPTS) * NPTS;
  const int wave = tid >> 5;
  const int lane = tid & 31;
  __syncthreads();

  // ---- stage 1: E = alpha_i - alpha_j + delta ; DV = val_j + delta
  for (int e = wave; e < 272; e += 8) {
    const int t    = e / KP1;
    const int slot = e - t * KP1;
    const int gi = base_t + t;
    const int gj = cbase + nbr[gi * KP1 + slot];
    const float dpx = pos[gi * 3 + 0] - pos[gj * 3 + 0];
    const float dpy = pos[gi * 3 + 1] - pos[gj * 3 + 1];
    const float dpz = pos[gi * 3 + 2] - pos[gj * 3 + 2];
#pragma unroll
    for (int h = 0; h < 2; ++h) {
      const int c = lane + 32 * h;
      const float delta = dpx * sWp[c] + dpy * sWp[64 + c] + dpz * sWp[128 + c] + sbp[c];
      sE [e * ES  + c] = Adst[(size_t)gi * CH + c] - Asrc[(size_t)gj * CH + c] + delta;
      sDV[e * DVS + c] = Vp[(size_t)gj * CH + c] + delta;
    }
  }
  __syncthreads();

  // ---- stage 2: S = E @ W_attn + b_attn (fp32 WMMA), overwrite E in-place.
  // Each wave owns disjoint M-tiles (rows), so in-place write is race-free.
  const int half = lane >> 4;
  const int l16  = lane & 15;
  for (int mt = wave; mt < 17; mt += 8) {
    v8f acc0 = {}, acc1 = {}, acc2 = {}, acc3 = {};
    const int arow = mt * 16 + l16;
#pragma unroll
    for (int kk = 0; kk < CH; kk += 4) {
      const int kb = kk + 2 * half;
      v2f a;  a.x = sE[arow * ES + kb];  a.y = sE[arow * ES + kb + 1];
      const int pr = (kb >> 1) * PS;
      const v2f b0 = *(const v2f*)&sWa[(pr + l16 +  0) * 2];
      const v2f b1 = *(const v2f*)&sWa[(pr + l16 + 16) * 2];
      const v2f b2 = *(const v2f*)&sWa[(pr + l16 + 32) * 2];
      const v2f b3 = *(const v2f*)&sWa[(pr + l16 + 48) * 2];
      acc0 = __builtin_amdgcn_wmma_f32_16x16x4_f32(false, a, false, b0, (short)0, acc0, false, false);
      acc1 = __builtin_amdgcn_wmma_f32_16x16x4_f32(false, a, false, b1, (short)0, acc1, false, false);
      acc2 = __builtin_amdgcn_wmma_f32_16x16x4_f32(false, a, false, b2, (short)0, acc2, false, false);
      acc3 = __builtin_amdgcn_wmma_f32_16x16x4_f32(false, a, false, b3, (short)0, acc3, false, false);
    }
#pragma unroll
    for (int v = 0; v < 8; ++v) {
      const int r = mt * 16 + v + 8 * half;
      sE[r * ES + l16 +  0] = acc0[v] + sba[l16 +  0];
      sE[r * ES + l16 + 16] = acc1[v] + sba[l16 + 16];
      sE[r * ES + l16 + 32] = acc2[v] + sba[l16 + 32];
      sE[r * ES + l16 + 48] = acc3[v] + sba[l16 + 48];
    }
  }
  __syncthreads();

  // ---- stage 3: per-(target,channel) softmax over 17 edges + weighted sum
  for (int p = tid; p < 16 * CH; p += 256) {
    const int t = p >> 6;
    const int c = p & 63;
    const int r0 = t * KP1;
    float m = -3.0e38f;
#pragma unroll
    for (int s = 0; s < KP1; ++s) m = fmaxf(m, sE[(r0 + s) * ES + c]);
    float z = 0.f, o = 0.f;
#pragma unroll
    for (int s = 0; s < KP1; ++s) {
      const float w = __expf(sE[(r0 + s) * ES + c] - m);
      z += w;
      o += w * sDV[(r0 + s) * DVS + c];
    }
    out[(size_t)(base_t + t) * CH + c] = o / z;
  }
}

// ---------------------------------------------------------------------------
extern "C" void kernel_launch(void* const* d_in, const int* in_sizes, int n_in,
                              void* d_out, int out_size, void* d_ws, size_t ws_size,
                              hipStream_t stream) {
  (void)in_sizes; (void)n_in; (void)out_size; (void)ws_size;
  const float* x      = (const float*)d_in[0];
  const float* pos    = (const float*)d_in[1];
  /* d_in[2] = batch (layout known: contiguous equal clouds) — unused */
  const float* W_pos  = (const float*)d_in[3];
  const float* b_pos  = (const float*)d_in[4];
  const float* W_attn = (const float*)d_in[5];
  const float* b_attn = (const float*)d_in[6];
  const float* W_val  = (const float*)d_in[7];
  const float* W_src  = (const float*)d_in[8];
  const float* W_dst  = (const float*)d_in[9];
  float* out = (float*)d_out;

  // workspace: nbr (int[32768*17], 2.23MB, 256B-aligned size) + 3 x 8MB fp32
  char* ws = (char*)d_ws;
  int*   nbr   = (int*)ws;
  float* a_dst = (float*)(ws + (size_t)BATCH * NPTS * KP1 * sizeof(int));
  float* a_src = a_dst + (size_t)BATCH * NPTS * CH;
  float* vproj = a_src + (size_t)BATCH * NPTS * CH;

  const size_t smem = (size_t)(272 * ES + 272 * DVS + 32 * PS * 2 + 192 + 64 + 64)
                      * sizeof(float);                     // 160,064 B
  hipFuncSetAttribute((const void*)edge_kernel,
                      hipFuncAttributeMaxDynamicSharedMemorySize, (int)smem);

  knn_kernel <<<dim3(BATCH * (NPTS / 256)), dim3(256), 0, stream>>>(pos, nbr);
  proj_kernel<<<dim3((BATCH * NPTS) / 128, 3), dim3(256), 0, stream>>>(
      x, W_dst, W_src, W_val, a_dst, a_src, vproj);
  edge_kernel<<<dim3((BATCH * NPTS) / 16), dim3(256), smem, stream>>>(
      pos, nbr, a_dst, a_src, vproj, W_pos, b_pos, W_attn, b_attn, out);
}